// mLSTMLayer_10780367913116
// MI455X (gfx1250) — compile-verified
//
#include <hip/hip_runtime.h>
#include <math.h>

typedef float v2f __attribute__((ext_vector_type(2)));
typedef float v8f __attribute__((ext_vector_type(8)));

#define NHEAD 8
#define DQH   128
#define DVH   256
#define BDIM  2
#define SDIM  2048
#define DDIM  2048
#define MROWS 4096   // B*S

// GEMM blocking
#define BM 128
#define BN 64
#define KB 16
#define LP 20        // padded LDS pitch (floats): 16B-aligned rows, conflict-free frags

#if __has_builtin(__builtin_amdgcn_global_load_async_to_lds_b128)
#define CDNA5_HAS_ASYNC 1
#else
#define CDNA5_HAS_ASYNC 0
#endif

#if __has_builtin(__builtin_amdgcn_s_wait_asynccnt)
#define WAITASYNC() __builtin_amdgcn_s_wait_asynccnt(0)
#else
#define WAITASYNC() asm volatile("s_wait_asynccnt 0x0" ::: "memory")
#endif

// Pointer element type must match the builtin's signature exactly:
// 'int __attribute__((vector_size(16)))' in AS1 (global) / AS3 (LDS).
typedef int v4i_ __attribute__((vector_size(4 * sizeof(int))));
typedef __attribute__((address_space(1))) v4i_ g_v4i;
typedef __attribute__((address_space(3))) v4i_ l_v4i;

// Stage one 16-byte vector global -> LDS (async DMA on CDNA5, fallback: reg copy)
__device__ __forceinline__ void lds_stage128(float* lptr, const float* gptr) {
#if CDNA5_HAS_ASYNC
  __builtin_amdgcn_global_load_async_to_lds_b128(
      (g_v4i*)gptr, (l_v4i*)lptr, 0, 0);
#else
  *(float4*)lptr = *(const float4*)gptr;
#endif
}

// ---------------------------------------------------------------------------
// WMMA GEMM:  Y[M,N] = X[M,K] * W[N,K]^T  (row-major fp32; M%128==0, N%64==0,
// K%16==0).  Block = 256 threads (8 waves) -> 128x64 tile; wave w owns rows
// [w*16, w*16+16).  K-chunks of 16 double-buffered in LDS via async-to-LDS.
// ---------------------------------------------------------------------------
__global__ __launch_bounds__(256)
void wmma_gemm_xwt(const float* __restrict__ X, const float* __restrict__ W,
                   float* __restrict__ Y, int M, int N, int K) {
  __shared__ float Xs[2][BM * LP];
  __shared__ float Ws[2][BN * LP];

  const int tid  = threadIdx.x;
  const int lane = tid & 31;
  const int wv   = tid >> 5;               // 0..7
  const int tilesN = N >> 6;
  const int bm = blockIdx.x / tilesN;
  const int bn = blockIdx.x - bm * tilesN;

  // global->LDS staging coords: X tile 128x16 = 512 float4 (2/thread),
  // W tile 64x16 = 256 float4 (1/thread)
  const int srow = tid >> 2;               // 0..63
  const int kc4  = (tid & 3) << 2;         // float4 column within K-chunk
  const float* Xg = X + (size_t)(bm * BM) * K;
  const float* Wg = W + (size_t)(bn * BN) * K;

  // WMMA fragment coords (A 16x4 / B 4x16): lanes 0-15 hold K={0,1}, 16-31 K={2,3}
  const int r  = lane & 15;
  const int kk = (lane >> 4) << 1;

  const int nch = K / KB;

#define STAGE_CHUNK(buf, k0)                                                   \
  do {                                                                         \
    lds_stage128(&Xs[buf][(srow)      * LP + kc4], Xg + (size_t)(srow)      * K + (k0) + kc4); \
    lds_stage128(&Xs[buf][(srow + 64) * LP + kc4], Xg + (size_t)(srow + 64) * K + (k0) + kc4); \
    lds_stage128(&Ws[buf][(srow)      * LP + kc4], Wg + (size_t)(srow)      * K + (k0) + kc4); \
  } while (0)

  v8f c0 = {}; v8f c1 = {}; v8f c2 = {}; v8f c3 = {};

  STAGE_CHUNK(0, 0);
  WAITASYNC();
  __syncthreads();

  for (int c = 0; c < nch; ++c) {
    const int cur = c & 1;
    if (c + 1 < nch) STAGE_CHUNK(1 - cur, (c + 1) * KB);   // overlap with compute

    const float* xb = &Xs[cur][(wv * 16 + r) * LP + kk];
    const float* w0 = &Ws[cur][(r)           * LP + kk];
    const float* w1 = &Ws[cur][(16 + r)      * LP + kk];
    const float* w2 = &Ws[cur][(32 + r)      * LP + kk];
    const float* w3 = &Ws[cur][(48 + r)      * LP + kk];
#pragma unroll
    for (int ks = 0; ks < KB / 4; ++ks) {
      v2f a  = *(const v2f*)(xb + ks * 4);
      v2f b0 = *(const v2f*)(w0 + ks * 4);
      v2f b1 = *(const v2f*)(w1 + ks * 4);
      v2f b2 = *(const v2f*)(w2 + ks * 4);
      v2f b3 = *(const v2f*)(w3 + ks * 4);
      c0 = __builtin_amdgcn_wmma_f32_16x16x4_f32(false, a, false, b0, (short)0, c0, false, false);
      c1 = __builtin_amdgcn_wmma_f32_16x16x4_f32(false, a, false, b1, (short)0, c1, false, false);
      c2 = __builtin_amdgcn_wmma_f32_16x16x4_f32(false, a, false, b2, (short)0, c2, false, false);
      c3 = __builtin_amdgcn_wmma_f32_16x16x4_f32(false, a, false, b3, (short)0, c3, false, false);
    }

    WAITASYNC();          // next chunk's DMA landed (per-wave)
    __syncthreads();      // visible to all waves
  }
#undef STAGE_CHUNK

  // C/D layout: VGPR j: lanes 0-15 -> M=j, lanes 16-31 -> M=j+8; N = lane&15
  const int mrow = bm * BM + wv * 16 + ((lane >> 4) << 3);
  const int ncol = bn * BN + (lane & 15);
  float* Yp = Y + (size_t)mrow * N + ncol;
#pragma unroll
  for (int j = 0; j < 8; ++j) {
    Yp[(size_t)j * N +  0] = c0[j];
    Yp[(size_t)j * N + 16] = c1[j];
    Yp[(size_t)j * N + 32] = c2[j];
    Yp[(size_t)j * N + 48] = c3[j];
  }
}

// ---------------------------------------------------------------------------
// Gate projections: i_pre/f_pre = soft_cap(x @ Wg^T + b).  One wave per row,
// all 8 heads per wave; wave32 butterfly reduction.
// ---------------------------------------------------------------------------
__device__ __forceinline__ float soft_cap15(float z) {
  return 15.0f * tanhf(z * (1.0f / 15.0f));
}

__global__ __launch_bounds__(256)
void gates_kernel(const float* __restrict__ x,
                  const float* __restrict__ Wig, const float* __restrict__ big,
                  const float* __restrict__ Wfg, const float* __restrict__ bfg,
                  float* __restrict__ ig, float* __restrict__ fg) {
  const int lane = threadIdx.x & 31;
  const int wave = threadIdx.x >> 5;
  const int m = blockIdx.x * 8 + wave;
  if (m >= MROWS) return;
  const float* xp = x + (size_t)m * DDIM;
  float aI[NHEAD] = {0,0,0,0,0,0,0,0};
  float aF[NHEAD] = {0,0,0,0,0,0,0,0};
  for (int k = lane; k < DDIM; k += 32) {
    float xv = xp[k];
#pragma unroll
    for (int g = 0; g < NHEAD; ++g) {
      aI[g] = fmaf(xv, Wig[(size_t)g * DDIM + k], aI[g]);
      aF[g] = fmaf(xv, Wfg[(size_t)g * DDIM + k], aF[g]);
    }
  }
#pragma unroll
  for (int off = 16; off >= 1; off >>= 1) {
#pragma unroll
    for (int g = 0; g < NHEAD; ++g) {
      aI[g] += __shfl_xor(aI[g], off, 32);
      aF[g] += __shfl_xor(aF[g], off, 32);
    }
  }
  if (lane == 0) {
#pragma unroll
    for (int g = 0; g < NHEAD; ++g) {
      ig[(size_t)m * NHEAD + g] = soft_cap15(aI[g] + big[g]);
      fg[(size_t)m * NHEAD + g] = soft_cap15(aF[g] + bfg[g]);
    }
  }
}

// ---------------------------------------------------------------------------
// mLSTM recurrent scan.  One block per (batch, head); 1024 threads (32 waves).
// Wave w owns DV columns [w*8, w*8+8); lane l owns DQ rows [l*4, l*4+4).
// State C (128x256 f32) fully VGPR-resident: 32 f32 per thread.
// q*C and q*n reductions are wave32 __shfl_xor butterflies.
// ---------------------------------------------------------------------------
__global__ __launch_bounds__(1024)
void scan_kernel(const float* __restrict__ q, const float* __restrict__ k,
                 const float* __restrict__ v,
                 const float* __restrict__ ig, const float* __restrict__ fg,
                 float* __restrict__ h, float* __restrict__ Cf,
                 float* __restrict__ nf, float* __restrict__ mf) {
  const int bh   = blockIdx.x;           // 0..B*NH-1
  const int b    = bh >> 3;
  const int hd   = bh & 7;
  const int lane = threadIdx.x & 31;     // d-group
  const int wave = threadIdx.x >> 5;     // v-group
  const int d0   = lane * 4;
  const int v0   = wave * 8;
  const float scale = 0.088388347648318447f;   // 128^-0.5

  float C[4][8] = {{0}};
  float n[4] = {0, 0, 0, 0};
  float m = 0.0f;

  for (int t = 0; t < SDIM; ++t) {
    const size_t row = (size_t)b * SDIM + t;
    const float* qp = q + (row * NHEAD + hd) * DQH;
    const float* kp = k + (row * NHEAD + hd) * DQH;
    const float* vp = v + (row * NHEAD + hd) * DVH;
    const float it = ig[row * NHEAD + hd];
    const float ft = fg[row * NHEAD + hd];

    // prefetch next timestep (global_prefetch_b8)
    if (t + 1 < SDIM) {
      __builtin_prefetch(qp + (size_t)NHEAD * DQH + d0, 0, 1);
      __builtin_prefetch(kp + (size_t)NHEAD * DQH + d0, 0, 1);
      __builtin_prefetch(vp + (size_t)NHEAD * DVH + v0, 0, 1);
    }

    // log_sigmoid(ft), numerically stable
    const float fl = fminf(ft, 0.0f) - log1pf(__expf(-fabsf(ft)));
    const float mn = fmaxf(fl + m, it);
    const float F  = __expf(fl + m - mn);
    const float I  = __expf(it - mn);

    float qs[4], ks[4], vv[8];
#pragma unroll
    for (int i = 0; i < 4; ++i) { qs[i] = qp[d0 + i] * scale; ks[i] = kp[d0 + i]; }
#pragma unroll
    for (int j = 0; j < 8; ++j) vv[j] = vp[v0 + j];

    float hp[8] = {0, 0, 0, 0, 0, 0, 0, 0};
#pragma unroll
    for (int i = 0; i < 4; ++i) {
      const float ik = I * ks[i];
#pragma unroll
      for (int j = 0; j < 8; ++j) {
        float c = fmaf(F, C[i][j], ik * vv[j]);
        C[i][j] = c;
        hp[j] = fmaf(qs[i], c, hp[j]);
      }
      n[i] = fmaf(F, n[i], ik);
    }
    float qn = fmaf(qs[0], n[0], fmaf(qs[1], n[1], fmaf(qs[2], n[2], qs[3] * n[3])));

#pragma unroll
    for (int off = 16; off >= 1; off >>= 1) {
      qn += __shfl_xor(qn, off, 32);
#pragma unroll
      for (int j = 0; j < 8; ++j) hp[j] += __shfl_xor(hp[j], off, 32);
    }

    m = mn;
    const float denom = fmaxf(fabsf(qn), __expf(-mn)) + 1e-6f;
    const float inv = 1.0f / denom;
    if (lane == 0) {
      float* ho = h + (row * NHEAD + hd) * DVH + v0;
#pragma unroll
      for (int j = 0; j < 8; ++j) ho[j] = hp[j] * inv;
    }
  }

  // final state: C (B,NH,DQ,DV), n (B,NH,DQ), m (B,NH)
  float* Cp = Cf + ((size_t)bh * DQH) * DVH;
#pragma unroll
  for (int i = 0; i < 4; ++i)
#pragma unroll
    for (int j = 0; j < 8; ++j)
      Cp[(size_t)(d0 + i) * DVH + v0 + j] = C[i][j];
  if (wave == 0) {
#pragma unroll
    for (int i = 0; i < 4; ++i) nf[(size_t)bh * DQH + d0 + i] = n[i];
  }
  if (threadIdx.x == 0) mf[bh] = m;
}

// ---------------------------------------------------------------------------
// Per-head layernorm * gamma, then sigmoid(og) gating.  One wave per
// (row, head): 256 elements, 8 per lane, wave32 butterfly mean/var.
// ---------------------------------------------------------------------------
__global__ __launch_bounds__(256)
void normgate_kernel(const float* __restrict__ h, const float* __restrict__ og,
                     const float* __restrict__ gamma, float* __restrict__ hout) {
  const int lane = threadIdx.x & 31;
  const int head = threadIdx.x >> 5;     // 0..7
  const size_t row = blockIdx.x;         // 0..MROWS-1
  const size_t base = row * DDIM + (size_t)head * DVH + lane * 8;
  const float* hp = h + base;            // (B,S,NH,DHV) is linear in (row, head*256+j)

  float vals[8];
  float s = 0.0f;
#pragma unroll
  for (int j = 0; j < 8; ++j) { vals[j] = hp[j]; s += vals[j]; }
#pragma unroll
  for (int off = 16; off >= 1; off >>= 1) s += __shfl_xor(s, off, 32);
  const float mu = s * (1.0f / DVH);

  float vs = 0.0f;
#pragma unroll
  for (int j = 0; j < 8; ++j) { float d = vals[j] - mu; vs = fmaf(d, d, vs); }
#pragma unroll
  for (int off = 16; off >= 1; off >>= 1) vs += __shfl_xor(vs, off, 32);
  const float rstd = rsqrtf(vs * (1.0f / DVH) + 1e-6f);

  const float* gp = gamma + (size_t)head * DVH + lane * 8;
  const float* op = og + base;
  float* out = hout + base;
#pragma unroll
  for (int j = 0; j < 8; ++j) {
    float hn = (vals[j] - mu) * rstd * gp[j];
    float sg = 1.0f / (1.0f + __expf(-op[j]));
    out[j] = sg * hn;
  }
}

// ---------------------------------------------------------------------------
extern "C" void kernel_launch(void* const* d_in, const int* in_sizes, int n_in,
                              void* d_out, int out_size, void* d_ws, size_t ws_size,
                              hipStream_t stream) {
  const float* x     = (const float*)d_in[0];
  const float* Wq    = (const float*)d_in[1];
  const float* Wk    = (const float*)d_in[2];
  const float* Wv    = (const float*)d_in[3];
  const float* Wog   = (const float*)d_in[4];
  const float* Wig   = (const float*)d_in[5];
  const float* big   = (const float*)d_in[6];
  const float* Wfg   = (const float*)d_in[7];
  const float* bfg   = (const float*)d_in[8];
  const float* gamma = (const float*)d_in[9];
  const float* Wout  = (const float*)d_in[10];

  float* ws   = (float*)d_ws;
  float* q    = ws;                            // 4096*1024 = 4,194,304
  float* kbuf = ws + (size_t)4194304;          // 4,194,304
  float* v    = ws + (size_t)8388608;          // 4096*2048 = 8,388,608
  float* og   = ws + (size_t)16777216;         // 8,388,608
  float* h    = ws + (size_t)25165824;         // 8,388,608
  float* ig   = ws + (size_t)33554432;         // 32,768
  float* fg   = ws + (size_t)33587200;         // 32,768   (total 33,619,968 f32)
  float* hout = ws;                            // reuse q+k space after the scan

  float* y  = (float*)d_out;                   // (B,S,D)   8,388,608
  float* Cf = y  + (size_t)8388608;            // (B,NH,DQ,DV) 524,288
  float* nf = Cf + (size_t)524288;             // (B,NH,DQ)    2,048
  float* mf = nf + (size_t)2048;               // (B,NH)       16

  // Projections (fp32 WMMA): grid = (M/128)*(N/64)
  wmma_gemm_xwt<<<512,  256, 0, stream>>>(x, Wq,  q,    MROWS, 1024, DDIM);
  wmma_gemm_xwt<<<512,  256, 0, stream>>>(x, Wk,  kbuf, MROWS, 1024, DDIM);
  wmma_gemm_xwt<<<1024, 256, 0, stream>>>(x, Wv,  v,    MROWS, 2048, DDIM);
  wmma_gemm_xwt<<<1024, 256, 0, stream>>>(x, Wog, og,   MROWS, 2048, DDIM);
  gates_kernel<<<512, 256, 0, stream>>>(x, Wig, big, Wfg, bfg, ig, fg);

  // Recurrent scan: one block per (b, head)
  scan_kernel<<<BDIM * NHEAD, 1024, 0, stream>>>(q, kbuf, v, ig, fg, h, Cf, nf, mf);

  // Per-head norm + output gate
  normgate_kernel<<<MROWS, 256, 0, stream>>>(h, og, gamma, hout);

  // Output projection
  wmma_gemm_xwt<<<1024, 256, 0, stream>>>(hout, Wout, y, MROWS, 2048, 2048);
}